// GMMLayer_61598420959738
// MI455X (gfx1250) — compile-verified
//
#include <hip/hip_runtime.h>
#include <hip/hip_bf16.h>
#include <math.h>

typedef __attribute__((ext_vector_type(2))) float    v2f;
typedef __attribute__((ext_vector_type(8))) float    v8f;
typedef __attribute__((ext_vector_type(4))) unsigned v4u;
typedef __attribute__((ext_vector_type(8))) unsigned v8u;

// Problem constants (fixed by the reference: b=8, k=20, w=h=128, d=16)
constexpr int BB = 8;
constexpr int KK = 20;
constexpr int DD = 16;
constexpr int WH = 128 * 128;     // 16384

// ---- stats kernel tiling -------------------------------------------------
constexpr int TP   = 256;         // pixels per tile
constexpr int ROWF = TP + 4;      // padded row: 260 dwords -> 16B-aligned rows,
                                  // bank offset 4*m (all 16 distinct) on transposed reads
constexpr int BUFF = DD * ROWF + TP;          // floats per (img+x) buffer = 4416
constexpr int STATS_LDS = (2 * BUFF + 256 + 16 + 1) * 4;   // double buffer + accumulators

// ---- posterior kernel LDS layout (floats) --------------------------------
constexpr int P_IMGS = 0;                      // [16][128]
constexpr int P_MS   = 2048;                   // [20][272] (rows padded to stride 17)
constexpr int P_MUS  = P_MS + 20 * 272;        // [20][16]
constexpr int P_COEF = P_MUS + 320;            // [20]
constexpr int POST_LDS = (P_COEF + 20 + 12) * 4;

// ---------------------------------------------------------------------------
// Kernel 1: per-(b,k) weighted sufficient statistics via f32 WMMA.
//   C_g[m][n] += sum_p w_p*img[m,p]*img[n,p]   (weighted Gram)
//   C_s[m][*] += sum_p w_p*img[m,p]            (A x ones)
//   C_w[*][*] += sum_p w_p                     (W x ones)
// Tiles are DMA'd into LDS with global_load_async_to_lds_b128, double-buffered
// so the next tile's DMA overlaps this tile's WMMA work (ASYNCcnt pipelining).
// ---------------------------------------------------------------------------
__global__ __launch_bounds__(256) void gmm_stats(
    const float* __restrict__ x, const float* __restrict__ img,
    float* __restrict__ sig_out, float* __restrict__ mu_out,
    float* __restrict__ alpha_out) {
  extern __shared__ float smem[];
  const unsigned gss = __builtin_amdgcn_groupstaticsize();

  const int bk = blockIdx.x;
  const int b  = bk / KK;
  const int t  = threadIdx.x;
  const int lane = t & 31, wid = t >> 5;
  const int half = lane >> 4, m = lane & 15;

  float* accG = smem + 2 * BUFF;   // [8][32]
  float* accS = accG + 256;        // [16]
  float* accW = accS + 16;         // [1]
  accG[wid * 32 + lane] = 0.f;
  if (t < 16) accS[t] = 0.f;
  if (t == 0) accW[0] = 0.f;

  const float* imgb = img + (size_t)b * DD * WH;
  const float* xbk  = x   + (size_t)bk * WH;

  // Issue one tile's async DMA into buffer `bufsel`.
  // Per-wave op counts: waves 0-1 issue 5 (4 img + 1 x), waves 2-7 issue 4.
  auto issue_tile = [&](int base, int bufsel) {
    const unsigned bufoff = gss + (unsigned)(bufsel * BUFF * 4);
    #pragma unroll
    for (int i = 0; i < 4; ++i) {
      int idx4 = t * 4 + i * 1024;               // 16B chunk index (floats)
      int d = idx4 >> 8, p = idx4 & (TP - 1);
      unsigned loff = bufoff + (unsigned)((d * ROWF + p) * 4);
      unsigned goff = (unsigned)((d * WH + base + p) * 4);
      asm volatile("global_load_async_to_lds_b128 %0, %1, %2"
                   :: "v"(loff), "v"(goff), "s"(imgb) : "memory");
    }
    if (t < 64) {                                // 256 floats of x = 64 chunks
      unsigned loff = bufoff + (unsigned)(DD * ROWF * 4) + (unsigned)(t * 16);
      unsigned goff = (unsigned)((base + t * 4) * 4);
      asm volatile("global_load_async_to_lds_b128 %0, %1, %2"
                   :: "v"(loff), "v"(goff), "s"(xbk) : "memory");
    }
  };

  v8f Cg = {}; v8f Cs = {}; v8f Cw = {};
  v2f ones; ones.x = 1.f; ones.y = 1.f;

  issue_tile(0, 0);
  constexpr int NT = WH / TP;                    // 64 tiles
  for (int ti = 0; ti < NT; ++ti) {
    if (ti + 1 < NT) {
      issue_tile((ti + 1) * TP, (ti + 1) & 1);   // overlap DMA with compute
      if (wid < 2) asm volatile("s_wait_asynccnt 0x5" ::: "memory");
      else         asm volatile("s_wait_asynccnt 0x4" ::: "memory");
    } else {
      asm volatile("s_wait_asynccnt 0x0" ::: "memory");
    }
    __syncthreads();                             // all waves' tile data visible

    const float* bi = smem + (ti & 1) * BUFF;
    const float* bx = bi + DD * ROWF;
    const int wbase = wid * (TP / 8);            // 32 pixels per wave
    #pragma unroll 4
    for (int c4 = 0; c4 < TP / 8; c4 += 4) {
      int p = wbase + c4 + 2 * half;             // low half: K=0,1  high half: K=2,3
      float w0 = bx[p], w1 = bx[p + 1];
      float i0 = bi[m * ROWF + p], i1 = bi[m * ROWF + p + 1];
      v2f A;  A.x  = w0 * i0; A.y  = w1 * i1;
      v2f Bv; Bv.x = i0;      Bv.y = i1;
      v2f Wv; Wv.x = w0;      Wv.y = w1;
      Cg = __builtin_amdgcn_wmma_f32_16x16x4_f32(false, A,  false, Bv,   (short)0, Cg, false, false);
      Cs = __builtin_amdgcn_wmma_f32_16x16x4_f32(false, A,  false, ones, (short)0, Cs, false, false);
      Cw = __builtin_amdgcn_wmma_f32_16x16x4_f32(false, Wv, false, ones, (short)0, Cw, false, false);
    }
    __syncthreads();                             // before the buffer is re-DMA'd
  }

  // cross-wave reduction (ds_add_f32)
  #pragma unroll
  for (int r = 0; r < 8; ++r) atomicAdd(&accG[r * 32 + lane], Cg[r]);
  if (m == 0) {                                  // column n==0 carries S_x[m]
    #pragma unroll
    for (int r = 0; r < 8; ++r) atomicAdd(&accS[r + 8 * half], Cs[r]);
    if (half == 0) atomicAdd(&accW[0], Cw[0]);
  }
  __syncthreads();

  // finalize: thread t maps to C element (r=t>>5, lane) -> (mm, nn)
  const float sx    = accW[0];
  const float denom = sx + 1e-5f;
  {
    int r  = t >> 5;
    int mm = r + 8 * half;
    int nn = m;
    float g    = accG[r * 32 + lane];
    float Sm   = accS[mm], Sn = accS[nn];
    float mu_m = Sm / denom, mu_n = Sn / denom;
    float s = (g - mu_m * Sn - mu_n * Sm + mu_m * mu_n * sx) / denom
              + (mm == nn ? 1e-3f : 0.f);
    sig_out[(size_t)bk * 256 + mm * 16 + nn] = s;
    if (r == 0 && half == 0) mu_out[bk * 16 + nn] = accS[nn] / denom;
    if (t == 0) alpha_out[bk] = sx / (float)WH;
  }
}

// ---------------------------------------------------------------------------
// Kernel 2: per-(b,k) 16x16 Cholesky, M = L^-1 (lower), logdet, coef.
// ---------------------------------------------------------------------------
__global__ void gmm_chol(const float* __restrict__ sig,
                         const float* __restrict__ alpha,
                         float* __restrict__ Mout, float* __restrict__ coef) {
  int bk = blockIdx.x * 32 + threadIdx.x;
  if (bk >= BB * KK) return;
  float A[16][16], L[16][16], Mi[16][16];
  const float* s = sig + (size_t)bk * 256;
  for (int i = 0; i < 16; ++i)
    for (int j = 0; j < 16; ++j) A[i][j] = s[i * 16 + j];

  float logdet = 0.f;
  for (int j = 0; j < 16; ++j) {
    float ss = A[j][j];
    for (int tt = 0; tt < j; ++tt) ss -= L[j][tt] * L[j][tt];
    float ljj = sqrtf(ss);
    L[j][j] = ljj;
    logdet += logf(ljj);
    float inv = 1.0f / ljj;
    for (int i = j + 1; i < 16; ++i) {
      float s2 = A[i][j];
      for (int tt = 0; tt < j; ++tt) s2 -= L[i][tt] * L[j][tt];
      L[i][j] = s2 * inv;
    }
  }
  for (int c = 0; c < 16; ++c) {
    for (int i = 0; i < 16; ++i) Mi[i][c] = 0.f;
    Mi[c][c] = 1.0f / L[c][c];
    for (int i = c + 1; i < 16; ++i) {
      float s2 = 0.f;
      for (int tt = c; tt < i; ++tt) s2 += L[i][tt] * Mi[tt][c];
      Mi[i][c] = -s2 / L[i][i];
    }
  }
  float* mo = Mout + (size_t)bk * 256;
  for (int i = 0; i < 16; ++i)
    for (int j = 0; j < 16; ++j) mo[i * 16 + j] = Mi[i][j];
  const float LOG2PI = 1.8378770664093453f;
  coef[bk] = logf(alpha[bk]) - 0.5f * 16.f * LOG2PI - logdet;
}

// ---------------------------------------------------------------------------
// Kernel 3: posterior. img tile (16 rows x 128 px) is fetched by ONE TDM
// tensor_load_to_lds (2D descriptor, data_size=4B, dim0 stride=WH) issued by
// wave 0, overlapping the per-class M/mu/coef staging by the other waves.
// Per wave/class: z = M_k x (img - mu_k) via 4 chained f32 WMMAs,
// maha = sum z^2 (shfl_xor 16), all 20 classes kept in registers.
// ---------------------------------------------------------------------------
__global__ __launch_bounds__(256) void gmm_post(
    const float* __restrict__ img, const float* __restrict__ Mw,
    const float* __restrict__ mu, const float* __restrict__ coef,
    float* __restrict__ out) {
  extern __shared__ float smem[];
  const unsigned gss = __builtin_amdgcn_groupstaticsize();
  float* imgs  = smem + P_IMGS;   // [16][128] contiguous rows (TDM layout)
  float* Ms    = smem + P_MS;     // [20][272], rows padded to stride 17
  float* mus   = smem + P_MUS;    // [20][16]
  float* coefs = smem + P_COEF;   // [20]

  const int b = blockIdx.y;
  const int px_base = blockIdx.x * 128;
  const int t = threadIdx.x;

  if (t < 32) {  // wave 0 kicks off the tensor DMA (EXEC-independent, per wave)
    unsigned long long ga =
        (unsigned long long)(img + (size_t)b * DD * WH + px_base);
    v4u g0; v8u g1;
    g0.x = 1u;                                        // count=1, user mode
    g0.y = gss + (unsigned)(P_IMGS * 4);              // lds_addr
    g0.z = (unsigned)ga;                              // global_addr[31:0]
    g0.w = (unsigned)((ga >> 32) & 0x1FFFFFFu) | (2u << 30);  // addr[56:32] | type=2
    g1.s0 = (2u << 16);                               // data_size=4B, no mask/flags
    g1.s1 = ((unsigned)WH & 0xFFFFu) << 16;           // tensor_dim0[15:0]
    g1.s2 = ((unsigned)WH >> 16) | (16u << 16);       // tensor_dim0[31:16] | tensor_dim1
    g1.s3 = (128u << 16);                             // tile_dim0 = 128
    g1.s4 = 16u;                                      // tile_dim1 = 16, tile_dim2 = 0
    g1.s5 = (unsigned)WH;                             // tensor_dim0_stride[31:0]
    g1.s6 = 0u;                                       // stride0[47:32] | stride1[15:0]
    g1.s7 = 0u;                                       // stride1[47:16]
    asm volatile("tensor_load_to_lds %0, %1" :: "s"(g0), "s"(g1) : "memory");
  }

  // Stage per-class tables with normal loads (overlaps the TDM transfer).
  const float* Mb = Mw + (size_t)b * KK * 256;
  for (int idx = t; idx < KK * 256; idx += 256) {
    int k = idx >> 8, j = idx & 255;
    Ms[k * 272 + (j >> 4) * 17 + (j & 15)] = Mb[idx];
  }
  for (int idx = t; idx < KK * 16; idx += 256)
    mus[(idx >> 4) * 16 + (idx & 15)] = mu[b * KK * 16 + idx];
  if (t < KK) coefs[t] = coef[b * KK + t];

  if (t < 32) __builtin_amdgcn_s_wait_tensorcnt(0);
  __syncthreads();

  const int lane = t & 31, wv = t >> 5;
  const int half = lane >> 4, mr = lane & 15;
  const int pw = wv * 16 + mr;            // pixel column n for this lane

  float sum = 1e-5f;
  float yk[KK];
  #pragma unroll
  for (int k = 0; k < KK; ++k) {
    v8f c = {};
    #pragma unroll
    for (int dc = 0; dc < 16; dc += 4) {
      int kkk = dc + 2 * half;
      v2f a;  a.x  = Ms[k * 272 + mr * 17 + kkk];
              a.y  = Ms[k * 272 + mr * 17 + kkk + 1];
      v2f bb; bb.x = imgs[kkk * 128 + pw]       - mus[k * 16 + kkk];
              bb.y = imgs[(kkk + 1) * 128 + pw] - mus[k * 16 + kkk + 1];
      c = __builtin_amdgcn_wmma_f32_16x16x4_f32(false, a, false, bb, (short)0, c, false, false);
    }
    float part = 0.f;
    #pragma unroll
    for (int r = 0; r < 8; ++r) part += c[r] * c[r];
    float maha = part + __shfl_xor(part, 16, 32);   // combine M rows 0-7 / 8-15
    float y = __expf(coefs[k] - 0.5f * maha);
    yk[k] = y;
    sum += y;
  }
  if (half == 0) {
    float inv = 1.0f / sum;               // sum already includes eps=1e-5
    int po = px_base + pw;
    #pragma unroll
    for (int k = 0; k < KK; ++k)
      out[((size_t)(b * KK + k)) * WH + po] = yk[k] * inv;
  }
}

// ---------------------------------------------------------------------------
extern "C" void kernel_launch(void* const* d_in, const int* in_sizes, int n_in,
                              void* d_out, int out_size, void* d_ws, size_t ws_size,
                              hipStream_t stream) {
  (void)in_sizes; (void)n_in; (void)out_size; (void)ws_size;
  const float* x   = (const float*)d_in[0];   // [8,20,128,128]
  const float* img = (const float*)d_in[1];   // [8,16,128,128]
  float* out = (float*)d_out;                 // [8,20,128,128]

  float* ws    = (float*)d_ws;
  float* sig   = ws;                          // 160*256
  float* Mw    = ws + 40960;                  // 160*256
  float* muw   = ws + 81920;                  // 160*16
  float* alpha = ws + 84480;                  // 160
  float* coef  = ws + 84640;                  // 160

  gmm_stats<<<BB * KK, 256, STATS_LDS, stream>>>(x, img, sig, muw, alpha);
  gmm_chol<<<(BB * KK + 31) / 32, 32, 0, stream>>>(sig, alpha, Mw, coef);
  gmm_post<<<dim3(WH / 128, BB), 256, POST_LDS, stream>>>(img, Mw, muw, coef, out);
}